// PaGNNConv_8607114461518
// MI455X (gfx1250) — compile-verified
//
#include <hip/hip_runtime.h>
#include <hip/hip_bf16.h>

// ---------------------------------------------------------------------------
// PaGNN partial-aggregation GCN conv for MI455X (gfx1250, wave32).
//
// Inputs (setup_inputs order):
//   d_in[0] x          [N,32]  f32
//   d_in[1] mask       [N,1]   f32
//   d_in[2] W          [32,32] f32   (out = ratio @ W.T + b)
//   d_in[3] b          [32]    f32
//   d_in[4] edge_index [2,E]   i32   (row = e[0:E], col = e[E:2E])
//
// Workspace layout (floats): dis[N] | sum_w[N] | den[N] | agg[N*32]  = 35N
//
// Roofline: all tensors (x 12.8MB, agg 12.8MB, dis/mask <1MB) fit in the
// 192MB L2, so the edge-scatter phase is L2-atomic-bound (51.2M f32 atomics).
// The final GEMM (100000x32 @ 32x32, 205 MFLOP) runs on the fp32 WMMA path
// (V_WMMA_F32_16X16X4_F32, 8 chained steps) at full reference precision.
// ---------------------------------------------------------------------------

typedef __attribute__((ext_vector_type(2))) float v2f;
typedef __attribute__((ext_vector_type(8))) float v8f;

__global__ void pagnn_zero_kernel(float* __restrict__ p, long n) {
    long i = (long)blockIdx.x * blockDim.x + threadIdx.x;
    long stride = (long)gridDim.x * blockDim.x;
    for (; i < n; i += stride) p[i] = 0.0f;
}

// One thread per edge: deg[row] += 1  (f32 hardware atomic).
__global__ void pagnn_deg_kernel(const int* __restrict__ rows,
                                 float* __restrict__ deg, int E) {
    int e = blockIdx.x * blockDim.x + threadIdx.x;
    if (e < E) unsafeAtomicAdd(deg + rows[e], 1.0f);
}

// In place: dis = deg > 0 ? rsqrt(deg) : 0
__global__ void pagnn_rsqrt_kernel(float* __restrict__ d, int N) {
    int n = blockIdx.x * blockDim.x + threadIdx.x;
    if (n < N) {
        float v = d[n];
        d[n] = (v > 0.0f) ? rsqrtf(v) : 0.0f;
    }
}

// One wave32 per edge; lane = feature index (0..31).
//   sum_w[row] += w                      (lane 0)
//   den[row]   += w*mask[col]            (lane 0)
//   agg[row,f] += w*mask[col]*x[col,f]   (all lanes, coalesced)
__global__ void pagnn_scatter_kernel(const float* __restrict__ x,
                                     const float* __restrict__ mask,
                                     const int* __restrict__ rows,
                                     const int* __restrict__ cols,
                                     const float* __restrict__ dis,
                                     float* __restrict__ sum_w,
                                     float* __restrict__ den,
                                     float* __restrict__ agg,
                                     int E) {
    int wave = (int)((blockIdx.x * blockDim.x + threadIdx.x) >> 5);
    if (wave >= E) return;
    int lane = threadIdx.x & 31;

    int r = rows[wave];
    int c = cols[wave];
    float w  = dis[r] * dis[c];
    float wm = w * mask[c];

    if (lane == 0) {
        if (w  != 0.0f) unsafeAtomicAdd(sum_w + r, w);
        if (wm != 0.0f) unsafeAtomicAdd(den + r, wm);
    }
    if (wm != 0.0f) {
        float xv = x[c * 32 + lane];
        xv = __builtin_isnan(xv) ? 0.0f : xv;   // nan_to_num(x)
        unsafeAtomicAdd(agg + r * 32 + lane, wm * xv);
    }
}

// One wave per 16x16 output tile. ratio[n,k] = scale[n]*agg[n,k],
// scale = den==0 ? 0 : sum_w/den. out = ratio @ W.T + b via
// V_WMMA_F32_16X16X4_F32, K=32 in 8 chained steps (full fp32 precision).
//
// f32 WMMA operand striping (ISA 7.12.2):
//   A (16x4): lanes 0-15 -> M=lane, VGPR0/1 = K=0/1; lanes 16-31 -> K=2/3
//   B (4x16): lanes 0-15 -> N=lane, VGPR0/1 = K=0/1; lanes 16-31 -> K=2/3
//   C/D (16x16): VGPR r -> M=r (lanes 0-15) / M=r+8 (lanes 16-31), N=lane&15
__global__ void pagnn_final_wmma_kernel(const float* __restrict__ agg,
                                        const float* __restrict__ sum_w,
                                        const float* __restrict__ den,
                                        const float* __restrict__ W,
                                        const float* __restrict__ bias,
                                        float* __restrict__ out,
                                        int N, int nTiles) {
    int wave = (int)((blockIdx.x * blockDim.x + threadIdx.x) >> 5);
    if (wave >= nTiles) return;          // wave-uniform: EXEC stays all-ones
    int lane = threadIdx.x & 31;

    int node_base = (wave >> 1) * 16;    // 16 nodes per tile-pair
    int ohalf     = (wave & 1) * 16;     // output columns 0-15 or 16-31
    int m   = lane & 15;                 // M row (A) or N col (B/C/D)
    int khi = (lane >> 4) << 1;          // +0 (lanes 0-15) or +2 (lanes 16-31)

    int node = node_base + m;
    bool in_range = node < N;
    int nodeLd = in_range ? node : (N - 1);

    float sw = sum_w[nodeLd];
    float dn = den[nodeLd];
    float scale = (in_range && dn != 0.0f) ? (sw / dn) : 0.0f;

    const float* aggRow = agg + (long)nodeLd * 32;
    const float* wRow   = W + (ohalf + m) * 32;   // B[k][o] = W[o*32+k]

    v8f c = {0.f, 0.f, 0.f, 0.f, 0.f, 0.f, 0.f, 0.f};
#pragma unroll
    for (int kk = 0; kk < 32; kk += 4) {
        int k = kk + khi;
        v2f a, b;
        a.x = scale * aggRow[k];
        a.y = scale * aggRow[k + 1];
        b.x = wRow[k];
        b.y = wRow[k + 1];
        // (neg_a, A, neg_b, B, c_mod, C, reuse_a, reuse_b)
        c = __builtin_amdgcn_wmma_f32_16x16x4_f32(
                false, a, false, b, (short)0, c, false, false);
    }

    float bb = bias[ohalf + m];
    int mrow = (lane >> 4) << 3;         // 0 or 8
    float* outCol = out + (long)(node_base + mrow) * 32 + ohalf + m;

    if (node_base + 16 <= N) {
        // Full tile (always true when N % 16 == 0): wave-uniform branch,
        // straight-line stores, no per-row EXEC manipulation.
#pragma unroll
        for (int r = 0; r < 8; ++r) {
            outCol[(long)r * 32] = c[r] + bb;
        }
    } else {
        // Tail tile: per-row guard.
#pragma unroll
        for (int r = 0; r < 8; ++r) {
            if (node_base + mrow + r < N) outCol[(long)r * 32] = c[r] + bb;
        }
    }
}

extern "C" void kernel_launch(void* const* d_in, const int* in_sizes, int n_in,
                              void* d_out, int out_size, void* d_ws, size_t ws_size,
                              hipStream_t stream) {
    const float* x    = (const float*)d_in[0];
    const float* mask = (const float*)d_in[1];
    const float* W    = (const float*)d_in[2];
    const float* bias = (const float*)d_in[3];
    const int* edges  = (const int*)d_in[4];

    int N = in_sizes[0] / 32;
    int E = in_sizes[4] / 2;
    const int* rows = edges;
    const int* cols = edges + E;

    float* ws    = (float*)d_ws;
    float* dis   = ws;            // deg, then rsqrt(deg) in place
    float* sum_w = ws + (long)N;
    float* den   = ws + 2L * N;
    float* agg   = ws + 3L * N;   // N*32

    long zn = 35L * N;
    pagnn_zero_kernel<<<1024, 256, 0, stream>>>(ws, zn);

    pagnn_deg_kernel<<<(E + 255) / 256, 256, 0, stream>>>(rows, dis, E);
    pagnn_rsqrt_kernel<<<(N + 255) / 256, 256, 0, stream>>>(dis, N);

    // one wave32 per edge -> 8 edges per 256-thread block
    long scatterThreads = (long)E * 32;
    int scatterBlocks = (int)((scatterThreads + 255) / 256);
    pagnn_scatter_kernel<<<scatterBlocks, 256, 0, stream>>>(
        x, mask, rows, cols, dis, sum_w, den, agg, E);

    int nTiles = ((N + 15) / 16) * 2;    // 16-node x 16-output tiles
    int finalBlocks = (nTiles * 32 + 255) / 256;
    pagnn_final_wmma_kernel<<<finalBlocks, 256, 0, stream>>>(
        agg, sum_w, den, W, bias, (float*)d_out, N, nTiles);
}